// MIL_fc_baens_wpr_90056874263226
// MI455X (gfx1250) — compile-verified
//
#include <hip/hip_runtime.h>
#include <math.h>

// ---------------- problem sizes ----------------
#define N_INST  50000
#define D_IN    1024
#define D_HID   512
#define D_ATT   256
#define N_ENS   4
#define CB_SIZE 256

// ---------------- tiling ----------------
#define BM    64                          // instances per block
#define NBLK  ((N_INST + BM - 1) / BM)    // 782
#define KC1   32                          // K-chunk for fc GEMM
#define KC2   32                          // K-chunk for attention GEMMs
#define NK1   (D_IN / KC1)                // 32 chunks
#define NK2   (D_HID / KC2)               // 16 chunks per ensemble

// padded LDS strides (floats) — avoid 64-bank conflicts under wave32 frag reads
#define XS_STRIDE 516
#define HS_STRIDE 36
#define WS_STRIDE 520
#define US_STRIDE 272

// LDS layout (float offsets); staging is double-buffered
#define XS_OFF     0
#define SB_OFF     (BM * XS_STRIDE)                    // 33024
#define SB_BUF     (BM * HS_STRIDE + KC1 * WS_STRIDE)  // 18944 (>= 2*KC2*US_STRIDE)
#define AROW2_OFF  (SB_OFF + 2 * SB_BUF)               // 70912
#define AS_OFF     (AROW2_OFF + 2 * BM)
#define SMEM_FLOATS (AS_OFF + BM)                      // 71104
#define SMEM_BYTES  (SMEM_FLOATS * sizeof(float))      // 284416 B (< 320 KB WGP)

typedef float v2f __attribute__((ext_vector_type(2)));
typedef float v8f __attribute__((ext_vector_type(8)));
typedef int   v4i_vs __attribute__((vector_size(16)));   // matches builtin param

#define AS1 __attribute__((address_space(1)))
#define AS3 __attribute__((address_space(3)))

// D = A(16x4, f32) * B(4x16, f32) + C(16x16, f32), exact fp32 matrix op.
static __device__ __forceinline__ v8f wmma_f32(v2f a, v2f b, v8f c) {
    return __builtin_amdgcn_wmma_f32_16x16x4_f32(
        /*neg_a=*/false, a, /*neg_b=*/false, b,
        /*c_mod=*/(short)0, c, /*reuse_a=*/false, /*reuse_b=*/false);
}

// ---- CDNA5 async global->LDS copy (ASYNCcnt), with safe fallback ----
#if __has_builtin(__builtin_amdgcn_global_load_async_to_lds_b128)
#define HAVE_ASYNC_LDS 1
#endif

static __device__ __forceinline__ void cp16_async(float* l, const float* g) {
#if defined(HAVE_ASYNC_LDS)
    __builtin_amdgcn_global_load_async_to_lds_b128(
        (AS1 v4i_vs*)(g),
        (AS3 v4i_vs*)(l),
        /*offset=*/0, /*cpol=*/0);
#else
    *(float4*)l = *(const float4*)g;
#endif
}

static __device__ __forceinline__ void async_wait0() {
#if defined(HAVE_ASYNC_LDS)
#if __has_builtin(__builtin_amdgcn_s_wait_asynccnt)
    __builtin_amdgcn_s_wait_asynccnt(0);
#else
    asm volatile("s_wait_asynccnt 0x0" ::: "memory");
#endif
#endif
}

// =====================================================================
// Kernel 1: fused per-block pipeline.
//  stage 1: x = relu(h @ fcW + b)           -> xs in LDS (64 x 512)
//  stage 2: per ensemble: tanh(x@Ua)*sigm(x@Ub) @ Uc -> per-row logit
//  stage 3: A_i = sigmoid(logit_i) -> d_out; deterministic partials of
//           sum_i A_i * x_i (512) and sum_i A_i into workspace.
// =====================================================================
__global__ void __launch_bounds__(256, 1)
mil_stage1(const float* __restrict__ h,   const float* __restrict__ fcW,
           const float* __restrict__ fcb, const float* __restrict__ Ua,
           const float* __restrict__ Ub,  const float* __restrict__ Uc,
           float* __restrict__ outA, float* __restrict__ partM,
           float* __restrict__ partS)
{
    extern __shared__ float smem[];
    float* xs    = smem + XS_OFF;
    float* sb    = smem + SB_OFF;      // two staging buffers of SB_BUF floats
    float* arow2 = smem + AROW2_OFF;   // [BM][2] per-row, per-col-half partials
    float* asig  = smem + AS_OFF;      // [BM] sigmoid(A)

    const int tid  = threadIdx.x;
    const int blk  = blockIdx.x;
    const int lane = tid & 31;
    const int w    = tid >> 5;         // wave 0..7
    const int half = lane >> 4;        // 0: lanes 0-15, 1: lanes 16-31
    const int lm   = lane & 15;
    const int row0 = blk * BM;

    const v8f vzero = {};

    // ---------------- Stage 1: x = relu(h @ fcW + b) ----------------
    const int mt    = w & 3;           // 16-row tile within the 64 rows
    const int nbase = (w >> 2) * 256;  // 256-col half of D_HID

    // staging thread mapping
    const int  hr  = tid >> 2;               // 0..63
    const int  hc  = (tid & 3) * 8;          // 0,8,16,24
    const int  hrg = row0 + hr;
    const bool hvalid = hrg < N_INST;
    const int  wr  = tid >> 3;               // 0..31
    const int  wc  = (tid & 7) * 64;

    v8f acc[16];
    #pragma unroll
    for (int j = 0; j < 16; ++j) acc[j] = vzero;

    auto stage1_issue = [&](int kc, float* buf) {
        const int kb = kc * KC1;
        float* hsb = buf;
        float* wsb = buf + BM * HS_STRIDE;
        if (hvalid) {
            const float* hsrc = h + (size_t)hrg * D_IN + kb + hc;
            cp16_async(hsb + hr * HS_STRIDE + hc,     hsrc);
            cp16_async(hsb + hr * HS_STRIDE + hc + 4, hsrc + 4);
        } else {
            float4 z; z.x = z.y = z.z = z.w = 0.0f;   // zero-pad tail rows
            *(float4*)(hsb + hr * HS_STRIDE + hc)     = z;
            *(float4*)(hsb + hr * HS_STRIDE + hc + 4) = z;
        }
        const float* wsrc = fcW + (size_t)(kb + wr) * D_HID + wc;
        float* wdst = wsb + wr * WS_STRIDE + wc;
        #pragma unroll
        for (int i = 0; i < 16; ++i) cp16_async(wdst + 4 * i, wsrc + 4 * i);
    };

    stage1_issue(0, sb);
    for (int kc = 0; kc < NK1; ++kc) {
        const int cur = kc & 1;
        float* buf = sb + cur * SB_BUF;
        async_wait0();                 // my async copies for chunk kc landed
        __syncthreads();               // everyone's landed; prev buf free
        if (kc + 1 < NK1) stage1_issue(kc + 1, sb + (cur ^ 1) * SB_BUF);
        if (hvalid && kc + 2 < NK1)    // L2 prefetch two chunks ahead
            __builtin_prefetch(h + (size_t)hrg * D_IN + (kc + 2) * KC1 + hc, 0, 1);

        float* hsL = buf;
        float* wsL = buf + BM * HS_STRIDE;
        #pragma unroll
        for (int k4 = 0; k4 < KC1 / 4; ++k4) {
            const int kk = k4 * 4 + 2 * half;   // A/B VGPR0 holds K={0,2}, VGPR1 K={1,3}
            v2f a;
            a.x = hsL[(mt * 16 + lm) * HS_STRIDE + kk];
            a.y = hsL[(mt * 16 + lm) * HS_STRIDE + kk + 1];
            #pragma unroll
            for (int j = 0; j < 16; ++j) {
                const int n = nbase + j * 16 + lm;
                v2f b;
                b.x = wsL[kk * WS_STRIDE + n];
                b.y = wsL[(kk + 1) * WS_STRIDE + n];
                acc[j] = wmma_f32(a, b, acc[j]);
            }
        }
        __syncthreads();               // done reading buf before it is reused
    }

    // epilogue: bias + relu -> xs   (C/D layout: VGPR r -> row r (+8 for hi half))
    #pragma unroll
    for (int j = 0; j < 16; ++j) {
        const int n = nbase + j * 16 + lm;
        const float bias = fcb[n];
        #pragma unroll
        for (int r = 0; r < 8; ++r) {
            float v = acc[j][r] + bias;
            xs[(mt * 16 + r + half * 8) * XS_STRIDE + n] = (v > 0.0f) ? v : 0.0f;
        }
    }
    __syncthreads();

    // ---------------- Stage 2: gated ensemble attention ----------------
    const int mt2 = w & 3;
    const int nq  = w >> 2;                  // 0..1 -> 128 attention cols each
    const int ur  = tid >> 3;                // 0..31
    const int uc  = (tid & 7) * 32;

    auto stage2_issue = [&](int q, float* buf) {   // q = e*NK2 + kc
        const int e  = q >> 4;
        const int kb = (q & (NK2 - 1)) * KC2;
        float* ua_b = buf;
        float* ub_b = buf + KC2 * US_STRIDE;
        const size_t base = (size_t)e * D_HID * D_ATT + (size_t)(kb + ur) * D_ATT + uc;
        #pragma unroll
        for (int i = 0; i < 8; ++i) {
            cp16_async(ua_b + ur * US_STRIDE + uc + 4 * i, Ua + base + 4 * i);
            cp16_async(ub_b + ur * US_STRIDE + uc + 4 * i, Ub + base + 4 * i);
        }
    };

    float rs[8];                             // per-row partial, summed over ensembles
    #pragma unroll
    for (int r = 0; r < 8; ++r) rs[r] = 0.0f;

    stage2_issue(0, sb);
    int q = 0;
    for (int e = 0; e < N_ENS; ++e) {
        v8f ta[8], tb[8];
        #pragma unroll
        for (int j = 0; j < 8; ++j) { ta[j] = vzero; tb[j] = vzero; }

        for (int kc = 0; kc < NK2; ++kc, ++q) {
            const int cur = q & 1;
            float* buf = sb + cur * SB_BUF;
            async_wait0();
            __syncthreads();
            if (q + 1 < N_ENS * NK2) stage2_issue(q + 1, sb + (cur ^ 1) * SB_BUF);

            float* uasL = buf;
            float* ubsL = buf + KC2 * US_STRIDE;
            const int kb = kc * KC2;
            #pragma unroll
            for (int k4 = 0; k4 < KC2 / 4; ++k4) {
                const int kk = k4 * 4 + 2 * half;
                v2f a;
                a.x = xs[(mt2 * 16 + lm) * XS_STRIDE + kb + kk];
                a.y = xs[(mt2 * 16 + lm) * XS_STRIDE + kb + kk + 1];
                #pragma unroll
                for (int j = 0; j < 8; ++j) {
                    const int n = nq * 128 + j * 16 + lm;
                    v2f b1, b2;
                    b1.x = uasL[kk * US_STRIDE + n];
                    b1.y = uasL[(kk + 1) * US_STRIDE + n];
                    b2.x = ubsL[kk * US_STRIDE + n];
                    b2.y = ubsL[(kk + 1) * US_STRIDE + n];
                    ta[j] = wmma_f32(a, b1, ta[j]);
                    tb[j] = wmma_f32(a, b2, tb[j]);
                }
            }
            __syncthreads();
        }

        // gate: tanh(t)*sigmoid(u), contracted with Uc column weights
        #pragma unroll
        for (int j = 0; j < 8; ++j) {
            const float ucv = Uc[e * D_ATT + nq * 128 + j * 16 + lm];
            #pragma unroll
            for (int r = 0; r < 8; ++r) {
                float g = tanhf(ta[j][r]) * (1.0f / (1.0f + expf(-tb[j][r])));
                rs[r] += g * ucv;
            }
        }
    }

    // deterministic lane reduction within each 16-lane half; unique LDS slots
    #pragma unroll
    for (int r = 0; r < 8; ++r) {
        float s = rs[r];
        s += __shfl_xor(s, 8, 16);
        s += __shfl_xor(s, 4, 16);
        s += __shfl_xor(s, 2, 16);
        s += __shfl_xor(s, 1, 16);
        if (lm == 0) arow2[(mt2 * 16 + r + half * 8) * 2 + nq] = s;
    }
    __syncthreads();

    // ---------------- Stage 3: A_i, partial A@x and sum(A) ----------------
    if (tid < BM) {
        const int rg = row0 + tid;
        const float logit = arow2[tid * 2] + arow2[tid * 2 + 1];
        const float a = (rg < N_INST) ? (1.0f / (1.0f + expf(-logit))) : 0.0f;
        asig[tid] = a;
        if (rg < N_INST) outA[8 + rg] = a;   // output #5 region: A[1,50000]
    }
    __syncthreads();

    #pragma unroll
    for (int cc = 0; cc < 2; ++cc) {
        const int c = tid + cc * 256;
        float s = 0.0f;
        for (int i = 0; i < BM; ++i) s += asig[i] * xs[i * XS_STRIDE + c];
        partM[(size_t)blk * D_HID + c] = s;
    }
    if (tid == 0) {
        float s = 0.0f;
        for (int i = 0; i < BM; ++i) s += asig[i];
        partS[blk] = s;
    }
}

// =====================================================================
// Kernel 2: single-block finalize. M, VQ (faithful +2*M.cb^T sign bug),
// classifier head, softmax, scalar outputs.
// =====================================================================
__global__ void __launch_bounds__(256)
mil_finalize(const float* __restrict__ partM, const float* __restrict__ partS,
             const float* __restrict__ codebook, const float* __restrict__ clsW,
             const float* __restrict__ clsb, float* __restrict__ out)
{
    __shared__ float Ms[D_HID];
    __shared__ float red[256];
    __shared__ int   redi[256];
    __shared__ float sh[4];     // 0: sumA, 1: vq_loss, 2..3: logits

    const int tid = threadIdx.x;

    // sum(A)
    {
        float s = 0.0f;
        for (int b = tid; b < NBLK; b += 256) s += partS[b];
        red[tid] = s;
    }
    __syncthreads();
    for (int st = 128; st > 0; st >>= 1) {
        if (tid < st) red[tid] += red[tid + st];
        __syncthreads();
    }
    if (tid == 0) sh[0] = red[0];
    __syncthreads();
    const float sumA = sh[0];

    // M = (A @ x) / sum(A)
    #pragma unroll
    for (int cc = 0; cc < 2; ++cc) {
        const int c = tid + cc * 256;
        float m = 0.0f;
        for (int b = 0; b < NBLK; ++b) m += partM[(size_t)b * D_HID + c];
        Ms[c] = m / sumA;
    }
    __syncthreads();

    // VQ argmin: dist_j = ||M||^2 + ||cb_j||^2 + 2 M.cb_j  (const term dropped)
    {
        float d = 0.0f;
        const float* cb = codebook + (size_t)tid * D_HID;
        for (int c = 0; c < D_HID; ++c) {
            const float v = cb[c];
            d += v * v + 2.0f * Ms[c] * v;
        }
        red[tid] = d; redi[tid] = tid;
    }
    __syncthreads();
    for (int st = 128; st > 0; st >>= 1) {
        if (tid < st) {
            const float d2 = red[tid + st]; const int i2 = redi[tid + st];
            if (d2 < red[tid] || (d2 == red[tid] && i2 < redi[tid])) {
                red[tid] = d2; redi[tid] = i2;
            }
        }
        __syncthreads();
    }
    const int bestIdx = redi[0];
    __syncthreads();

    // vq_loss = (1 + BETA) * mean((q - M)^2), BETA = 0.25
    {
        float v = 0.0f;
        #pragma unroll
        for (int cc = 0; cc < 2; ++cc) {
            const int c = tid + cc * 256;
            const float d = codebook[(size_t)bestIdx * D_HID + c] - Ms[c];
            v += d * d;
        }
        red[tid] = v;
    }
    __syncthreads();
    for (int st = 128; st > 0; st >>= 1) {
        if (tid < st) red[tid] += red[tid + st];
        __syncthreads();
    }
    if (tid == 0) sh[1] = 1.25f * (red[0] / (float)D_HID);

    // logits (2 classes)
    if (tid < 2) {
        float l = clsb[tid];
        for (int c = 0; c < D_HID; ++c) l += Ms[c] * clsW[c * 2 + tid];
        sh[2 + tid] = l;
    }
    __syncthreads();

    if (tid == 0) {
        const float l0 = sh[2], l1 = sh[3];
        const float mx = fmaxf(l0, l1);
        float p0 = expf(l0 - mx), p1 = expf(l1 - mx);
        const float ps = p0 + p1;
        p0 /= ps; p1 /= ps;
        out[0] = l0;  out[1] = l1;         // top_instance (top_k over 1 row -> row 0)
        out[2] = p0;  out[3] = p1;         // Y_prob
        out[4] = (l1 > l0) ? 1.0f : 0.0f;  // Y_hat (argmax, first-max tie break)
        out[5] = sh[1];                    // vq_loss
        out[6] = p0;  out[7] = p1;         // y_probs
    }
}

// =====================================================================
extern "C" void kernel_launch(void* const* d_in, const int* in_sizes, int n_in,
                              void* d_out, int out_size, void* d_ws, size_t ws_size,
                              hipStream_t stream) {
    (void)in_sizes; (void)n_in; (void)out_size; (void)ws_size;
    const float* h    = (const float*)d_in[0];
    const float* fcW  = (const float*)d_in[1];
    const float* fcb  = (const float*)d_in[2];
    const float* Ua   = (const float*)d_in[3];
    const float* Ub   = (const float*)d_in[4];
    const float* Uc   = (const float*)d_in[5];
    const float* clsW = (const float*)d_in[6];
    const float* clsb = (const float*)d_in[7];
    const float* cbk  = (const float*)d_in[8];
    float* out = (float*)d_out;

    float* partM = (float*)d_ws;                        // [NBLK][512]
    float* partS = partM + (size_t)NBLK * D_HID;        // [NBLK]

    // allow > 64KB dynamic LDS (WGP has 320KB); idempotent, not a stream op
    hipError_t err_ = hipFuncSetAttribute((const void*)mil_stage1,
                                          hipFuncAttributeMaxDynamicSharedMemorySize,
                                          (int)SMEM_BYTES);
    (void)err_;

    hipLaunchKernelGGL(mil_stage1, dim3(NBLK), dim3(256), SMEM_BYTES, stream,
                       h, fcW, fcb, Ua, Ub, Uc, out, partM, partS);
    hipLaunchKernelGGL(mil_finalize, dim3(1), dim3(256), 0, stream,
                       partM, partS, cbk, clsW, clsb, out);
}